// SBNetConvOperator_4827543240900
// MI455X (gfx1250) — compile-verified
//
#include <hip/hip_runtime.h>

// SBNet block-sparse 3x3 conv (stride 1, 'same') for gfx1250 / MI455X.
// FP32 implicit GEMM per 14x14 output block using V_WMMA_F32_16X16X4_F32.
//
// Each wave owns ONE M-tile (mtile = wave & 3) and up to 7 N-tiles: one
// A-fragment global_load_b64 feeds 7 independent WMMAs. The channel-step
// loop stays ROLLED (8 iters) to bound register pressure: per iteration a
// 9-load A clause (immediate offsets) then 63 ds+wmma pairs with DS
// immediate offsets off 7 per-tile base registers. K is tap-major
// (k' = t*64 + c); weights pre-repacked into d_ws fragment-major. Pad
// columns are address-clamped only: a B column affects only the same D
// column, which is never stored.

typedef __attribute__((ext_vector_type(2))) float v2f;
typedef __attribute__((ext_vector_type(8))) float v8f;

#define HH     448
#define CIN    64
#define NCOUT  64
#define BS     14          // output block stride
#define NPOS   196         // 14*14 output positions per block
#define KTOT   576         // 64 channels * 9 taps
#define CCHUNK 32          // channels staged per LDS pass
#define NKSTEP 144         // KTOT / 4

// ---- repack W[co][c][3][3] -> wp[(mtile*144 + kstep)*32 + lane][2] ----
// Lane l<16 supplies A rows m = mtile*16+l, k' = kstep*4 + {0,1};
// lane l>=16 supplies m = mtile*16+l-16, k' = kstep*4 + {2,3}.  k' = t*64+c.
__global__ __launch_bounds__(256)
void repack_weights(const float* __restrict__ w, float* __restrict__ wp)
{
    int idx = blockIdx.x * 256 + threadIdx.x;       // 0 .. 18431
    if (idx >= 4 * NKSTEP * 32) return;
    int lane  = idx & 31;
    int kstep = (idx >> 5) % NKSTEP;
    int mtile = (idx >> 5) / NKSTEP;
    int m  = mtile * 16 + (lane & 15);
    int kp = kstep * 4 + ((lane >> 4) << 1);
    #pragma unroll
    for (int i = 0; i < 2; ++i) {
        int kpi = kp + i;
        int t = kpi >> 6;          // tap 0..8
        int c = kpi & 63;          // channel
        wp[idx * 2 + i] = w[m * KTOT + c * 9 + t];
    }
}

__global__ __launch_bounds__(256)
void sbnet_conv_wmma_f32(const float* __restrict__ x,
                         const float* __restrict__ mask,
                         const float* __restrict__ wp,   // repacked weights
                         const float* __restrict__ bias,
                         float* __restrict__ out)
{
    __shared__ float lds_patch[CCHUNK * 256];   // [c_local][16x16] = 32 KB
    __shared__ int   s_active;

    const int tid = threadIdx.x;
    const int bx  = blockIdx.x;
    const int by  = blockIdx.y;
    const int img = blockIdx.z;
    const int x0  = bx * BS;
    const int y0  = by * BS;

    // ---- reduce_mask: max over the 16x16 gather window (offset -1) ----
    if (tid == 0) s_active = 0;
    __syncthreads();
    {
        int wy = y0 - 1 + (tid >> 4);
        int wx = x0 - 1 + (tid & 15);
        if (wy >= 0 && wy < HH && wx >= 0 && wx < HH) {
            float m = mask[(size_t)img * HH * HH + (size_t)wy * HH + wx];
            if (m > 0.5f) s_active = 1;   // benign race: all writers store 1
        }
    }
    __syncthreads();

    const size_t out_img_base = (size_t)img * NCOUT * HH * HH;

    if (!s_active) {
        for (int idx = tid; idx < NCOUT * NPOS; idx += 256) {
            int co = idx / NPOS;
            int p  = idx - co * NPOS;
            int py = p / BS, px = p - py * BS;
            out[out_img_base + (size_t)co * HH * HH +
                (size_t)(y0 + py) * HH + (x0 + px)] = 0.0f;
        }
        return;
    }

    const int wave   = tid >> 5;    // 0..7, uniform per wave
    const int lane   = tid & 31;
    const int lanelo = lane & 15;
    const int lanehi = lane >> 4;
    const int mtile  = wave & 3;    // every tile of this wave has this M-tile

    // Per-tile B base offsets (independent of chunk/tap): 7 registers.
    // Out-of-range tiles/columns are clamped -- their D columns never stored.
    int boffs[7];
    #pragma unroll
    for (int ti = 0; ti < 7; ++ti) {
        int tile = ti * 8 + wave;
        if (tile > 51) tile = 51;            // waves 4..7 lack a 7th tile
        int n0 = (tile >> 2) << 4;
        int n  = n0 + lanelo;
        int ncl = (n < NPOS) ? n : (NPOS - 1);
        int py = ncl / BS, px = ncl - py * BS;
        boffs[ti] = py * 16 + px + (lanehi << 9);
    }

    v8f acc[7] = {};

    for (int cb = 0; cb < 2; ++cb) {
        // ---- stage 32 channels of the 16x16 halo patch into LDS ----
        for (int idx = tid; idx < CCHUNK * 256; idx += 256) {
            int c  = cb * CCHUNK + (idx >> 8);
            int pp = idx & 255;
            int iy = y0 - 1 + (pp >> 4);
            int ix = x0 - 1 + (pp & 15);
            float v = 0.0f;
            if (iy >= 0 && iy < HH && ix >= 0 && ix < HH)
                v = x[((size_t)(img * CIN + c) * HH + iy) * HH + ix];
            lds_patch[idx] = v;
        }
        __syncthreads();

        // A fragment stream for this wave's M-tile (coalesced 256B/wave)
        const float* ap = wp + mtile * (NKSTEP * 64) + cb * 8 * 64 + (lane << 1);
        int csoff = 0;

        #pragma unroll 1
        for (int cs = 0; cs < 8; ++cs) {     // 4 channels per WMMA step
            // 9 tap A-fragments: one clause of imm-offset b64 loads
            v2f a[9];
            #pragma unroll
            for (int t = 0; t < 9; ++t)
                a[t] = *(const v2f*)(ap + t * (16 * 64));   // +4KB per tap
            #pragma unroll
            for (int t = 0; t < 9; ++t) {
                const int toff = (t / 3) * 16 + (t % 3);    // folds into DS imm
                #pragma unroll
                for (int ti = 0; ti < 7; ++ti) {
                    int baddr = boffs[ti] + csoff;          // CSE'd per tile
                    v2f b;
                    b.x = lds_patch[baddr + toff];
                    b.y = lds_patch[baddr + toff + 256];
                    acc[ti] = __builtin_amdgcn_wmma_f32_16x16x4_f32(
                        false, a[t], false, b, (short)0, acc[ti], false, false);
                }
            }
            ap    += 64;     // next 4 channels: +256B in A stream
            csoff += 1024;   // next 4 channels: +4 rows of 256 floats in LDS
        }
        __syncthreads();
    }

    // ---- store: D layout: M = vgpr + 8*(lane>=16), N = lane&15 ----
    const int m0 = mtile << 4;
    #pragma unroll
    for (int ti = 0; ti < 7; ++ti) {
        int tile = ti * 8 + wave;
        if (tile < 52) {
            int n0 = (tile >> 2) << 4;
            int n  = n0 + lanelo;
            if (n < NPOS) {
                int py = n / BS, px = n - py * BS;
                size_t pbase = out_img_base + (size_t)(y0 + py) * HH + (x0 + px);
                #pragma unroll
                for (int j = 0; j < 8; ++j) {
                    int co = m0 + j + (lanehi << 3);
                    out[pbase + (size_t)co * HH * HH] = acc[ti][j] + bias[co];
                }
            }
        }
    }
}

extern "C" void kernel_launch(void* const* d_in, const int* in_sizes, int n_in,
                              void* d_out, int out_size, void* d_ws, size_t ws_size,
                              hipStream_t stream) {
    const float* x      = (const float*)d_in[0];   // [4,64,448,448]
    const float* mask   = (const float*)d_in[1];   // [4,1,448,448]
    const float* weight = (const float*)d_in[2];   // [64,64,3,3]
    const float* bias   = (const float*)d_in[3];   // [64]
    float* out = (float*)d_out;                    // [4,64,448,448]
    float* wp  = (float*)d_ws;                     // 36864 floats = 147456 B

    // 1) repack weights into WMMA-fragment-major layout (stream-ordered)
    repack_weights<<<dim3((4 * NKSTEP * 32 + 255) / 256), dim3(256), 0, stream>>>(
        weight, wp);

    // 2) block-sparse implicit-GEMM conv
    dim3 grid(HH / BS, HH / BS, 4);                // 32 x 32 x 4 blocks
    dim3 block(256);                               // 8 wave32
    sbnet_conv_wmma_f32<<<grid, block, 0, stream>>>(x, mask, wp, bias, out);
}